// TripletLoss_71373766525718
// MI455X (gfx1250) — compile-verified
//
#include <hip/hip_runtime.h>

typedef __attribute__((ext_vector_type(2))) float v2f;
typedef __attribute__((ext_vector_type(4))) float v4f;
typedef __attribute__((ext_vector_type(8))) float v8f;

#define NROWS 8192
#define DIM   128
#define MARGINF 0.3f
#define EPSF   1e-12f

// WGP-scope prefetch (scope field 0 = WGP: pulls into all cache levels,
// including WGP$ -- the builtin only emits SCOPE_SE which stops at GL2).
// Addresses must be valid: WGP-scope prefetches are non-speculative.
__device__ __forceinline__ void prefetch_wgp(const void* p) {
    asm volatile("global_prefetch_b8 %0, off" :: "v"(p));
}

// ---------------- Kernel 1: per-row squared L2 norms (wave per row) ----------
__global__ void __launch_bounds__(256)
sqnorm_kernel(const float* __restrict__ X, float* __restrict__ sq) {
    const int lane = threadIdx.x & 31;
    const int wave = threadIdx.x >> 5;
    const int row  = blockIdx.x * 8 + wave;
    v4f x = *(const v4f*)(X + (size_t)row * DIM + lane * 4);
    float s = x.x * x.x + x.y * x.y + x.z * x.z + x.w * x.w;
#pragma unroll
    for (int m = 16; m >= 1; m >>= 1) s += __shfl_xor(s, m, 32);
    if (lane == 0) sq[row] = s;
}

// ---------------- Kernel 2: fused Gram-tile WMMA + hard mining ---------------
// Block = 256 threads (8 waves). Block owns rows [16b, 16b+16). Each wave
// walks column tiles jt = wave, wave+8, ... (64 tiles each). Per tile:
// 16x16 Gram block = sum of 32 V_WMMA_F32_16X16X4_F32 over K=128, then
// dist = sqrt(max(rsq + csq - 2g, eps)) and masked max/min accumulation.
//
// K-permutation trick: the reduction over K is order-independent, so we remap
// K identically for A and B fragments to make per-lane data contiguous.
// Lane loads float4 at k0 = 8t + 4*hi (hi = lane>>4). Virtual chunk 2t uses
// elements {0,1}, chunk 2t+1 uses elements {2,3}; union over both halves
// covers k = 8t..8t+7 exactly once -> 16 b128 loads per tile.
//
// Latency: next tile's rows (512B each = 4 lines) are prefetched at WGP scope
// while the current tile's WMMA chain runs, so the per-use b128 loads hit WGP$.
__global__ void __launch_bounds__(256)
triplet_tile_kernel(const float* __restrict__ X, const int* __restrict__ tgt,
                    const float* __restrict__ sq, float* __restrict__ hinge) {
    __shared__ float s_ap[8][16];
    __shared__ float s_an[8][16];

    const int lane  = threadIdx.x & 31;
    const int wave  = threadIdx.x >> 5;
    const int rsel  = lane & 15;   // row/col index within 16-wide tile
    const int hi    = lane >> 4;   // K-half select
    const int koff  = hi * 4;      // per-lane K base inside each 8-wide group
    const int ibase = blockIdx.x * 16;
    const float INF = __builtin_inff();

    // Resident A fragments for this block's 16-row strip (64 VGPRs as 16 quads).
    v4f afragq[16];
    const float* arow = X + (size_t)(ibase + rsel) * DIM + koff;
#pragma unroll
    for (int t = 0; t < 16; ++t)
        afragq[t] = *(const v4f*)(arow + t * 8);

    // Rows this lane's C-fragment elements belong to (M = v + 8*hi).
    float rsq[8];
    int   rtgt[8];
#pragma unroll
    for (int v = 0; v < 8; ++v) {
        const int row = ibase + v + 8 * hi;
        rsq[v]  = sq[row];
        rtgt[v] = tgt[row];
    }

    float ap[8], an[8];
#pragma unroll
    for (int v = 0; v < 8; ++v) { ap[v] = -INF; an[v] = INF; }

    // Prime WGP$ with this wave's first tile.
    {
        const float* prow = X + (size_t)(wave * 16 + rsel) * DIM;
        prefetch_wgp(prow);
        prefetch_wgp(prow + 32);
        prefetch_wgp(prow + 64);
        prefetch_wgp(prow + 96);
    }

    for (int jt = wave; jt < NROWS / 16; jt += 8) {
        const int jbase = jt * 16;
        const float* brow = X + (size_t)(jbase + rsel) * DIM + koff;

        // Prefetch next tile (clamped on last iteration; wave-uniform select,
        // so the address is always a valid row -- required for WGP scope).
        {
            const int pjt = (jt + 8 < NROWS / 16) ? (jt + 8) : jt;
            const float* prow = X + (size_t)(pjt * 16 + rsel) * DIM;
            prefetch_wgp(prow);
            prefetch_wgp(prow + 32);
            prefetch_wgp(prow + 64);
            prefetch_wgp(prow + 96);
        }

        v4f bfrag[16];
#pragma unroll
        for (int t = 0; t < 16; ++t)
            bfrag[t] = *(const v4f*)(brow + t * 8);

        v8f acc0 = {};
        v8f acc1 = {};
#pragma unroll
        for (int t = 0; t < 16; ++t) {
            v2f a0 = __builtin_shufflevector(afragq[t], afragq[t], 0, 1);
            v2f a1 = __builtin_shufflevector(afragq[t], afragq[t], 2, 3);
            v2f b0 = __builtin_shufflevector(bfrag[t],  bfrag[t],  0, 1);
            v2f b1 = __builtin_shufflevector(bfrag[t],  bfrag[t],  2, 3);
            acc0 = __builtin_amdgcn_wmma_f32_16x16x4_f32(
                false, a0, false, b0, (short)0, acc0, false, false);
            acc1 = __builtin_amdgcn_wmma_f32_16x16x4_f32(
                false, a1, false, b1, (short)0, acc1, false, false);
        }
        v8f g = acc0 + acc1;

        const float csq  = sq[jbase + rsel];
        const int   ctgt = tgt[jbase + rsel];
#pragma unroll
        for (int v = 0; v < 8; ++v) {
            const float d2 = rsq[v] + csq - 2.0f * g[v];
            const float d  = __builtin_sqrtf(fmaxf(d2, EPSF));
            if (rtgt[v] == ctgt) ap[v] = fmaxf(ap[v], d);
            else                 an[v] = fminf(an[v], d);
        }
    }

    // Reduce across the 16 lanes of each half (columns) -> per-row results.
#pragma unroll
    for (int m = 8; m >= 1; m >>= 1) {
#pragma unroll
        for (int v = 0; v < 8; ++v) {
            ap[v] = fmaxf(ap[v], __shfl_xor(ap[v], m, 16));
            an[v] = fminf(an[v], __shfl_xor(an[v], m, 16));
        }
    }
    if (rsel == 0) {
#pragma unroll
        for (int v = 0; v < 8; ++v) {
            s_ap[wave][v + 8 * hi] = ap[v];
            s_an[wave][v + 8 * hi] = an[v];
        }
    }
    __syncthreads();

    // Cross-wave reduce + hinge for this block's 16 rows.
    if (threadIdx.x < 16) {
        float bap = -INF, ban = INF;
#pragma unroll
        for (int w = 0; w < 8; ++w) {
            bap = fmaxf(bap, s_ap[w][threadIdx.x]);
            ban = fminf(ban, s_an[w][threadIdx.x]);
        }
        hinge[ibase + threadIdx.x] = fmaxf(bap - ban + MARGINF, 0.0f);
    }
}

// ---------------- Kernel 3: deterministic mean over 8192 hinges -------------
__global__ void __launch_bounds__(256)
reduce_kernel(const float* __restrict__ hinge, float* __restrict__ out) {
    __shared__ float buf[256];
    float s = 0.0f;
    for (int i = threadIdx.x; i < NROWS; i += 256) s += hinge[i];
    buf[threadIdx.x] = s;
    __syncthreads();
    for (int m = 128; m >= 1; m >>= 1) {
        if ((int)threadIdx.x < m) buf[threadIdx.x] += buf[threadIdx.x + m];
        __syncthreads();
    }
    if (threadIdx.x == 0) out[0] = buf[0] * (1.0f / (float)NROWS);  // LOSS_WEIGHT = 1
}

extern "C" void kernel_launch(void* const* d_in, const int* in_sizes, int n_in,
                              void* d_out, int out_size, void* d_ws, size_t ws_size,
                              hipStream_t stream) {
    (void)in_sizes; (void)n_in; (void)out_size; (void)ws_size;
    const float* X   = (const float*)d_in[0];
    const int*   tgt = (const int*)d_in[1];
    float* sq    = (float*)d_ws;          // 8192 floats
    float* hinge = sq + NROWS;            // 8192 floats
    float* out   = (float*)d_out;

    sqnorm_kernel<<<NROWS / 8, 256, 0, stream>>>(X, sq);
    triplet_tile_kernel<<<NROWS / 16, 256, 0, stream>>>(X, tgt, sq, hinge);
    reduce_kernel<<<1, 256, 0, stream>>>(hinge, out);
}